// Down_83674552861284
// MI455X (gfx1250) — compile-verified
//
#include <hip/hip_runtime.h>

#define K_OFF     8
#define P_PAIRS   125000
#define CH_IN     64
#define CH_OUT    128
#define N_OUT_C   250000
#define BN_EPS    1e-5f

typedef float v2f __attribute__((ext_vector_type(2)));
typedef float v8f __attribute__((ext_vector_type(8)));

// ---------------------------------------------------------------------------
// Kernel 1: gather -> f32 WMMA (16x16x4) -> atomic scatter-add
// grid = (ceil(P/128), K), block = 256 threads (8 waves; wave w owns output
// columns [w*16, w*16+16) ).
// ---------------------------------------------------------------------------
__global__ __launch_bounds__(256) void spconv_gemm(
    const float* __restrict__ feat,     // [N_IN, 64]
    const float* __restrict__ weight,   // [K, 64, 128]
    const int*   __restrict__ in_idx,   // [K, P]
    const int*   __restrict__ out_idx,  // [K, P]
    float*       __restrict__ out)      // [N_OUT, 128] accumulator
{
    // padded stride 68: A-frag read banks = (4*m + kk) % 64 -> conflict-free
    __shared__ float sA[16][68];
    __shared__ alignas(16) int sOut[16]; // out row per tile pair, -1 = invalid

    const int k       = blockIdx.y;
    const int pBase   = blockIdx.x * 128;
    const int tid     = threadIdx.x;
    const int lane    = tid & 31;
    const int wave    = tid >> 5;
    const int n0      = wave * 16;
    const int halfsel = lane >> 4;       // 0: lanes 0-15, 1: lanes 16-31
    const int m       = lane & 15;
    const int krow    = halfsel * 2;
    const bool full   = (pBase + 128) <= P_PAIRS;   // block-uniform

    // Preload B fragments for this wave's 16 columns; kept in VGPRs across all
    // 8 row tiles of this block (weight read once per 128 pairs).
    // V_WMMA_F32_16X16X4_F32 B layout: VGPR0 lanes0-15 K=kk+0 / lanes16-31
    // K=kk+2 ; VGPR1: K=kk+1 / K=kk+3 ; N = n0 + (lane&15).
    const float* __restrict__ Wk = weight + (size_t)k * CH_IN * CH_OUT;
    v2f bfrag[16];
#pragma unroll
    for (int s = 0; s < 16; ++s) {
        bfrag[s].x = Wk[(4 * s + krow    ) * CH_OUT + n0 + m];
        bfrag[s].y = Wk[(4 * s + krow + 1) * CH_OUT + n0 + m];
    }

    const int r  = tid >> 4;          // staging row 0..15
    const int c4 = (tid & 15) * 4;    // staging column group
    const unsigned colofs = (unsigned)(n0 + m);

    for (int t = 0; t < 8; ++t) {
        const int p0 = pBase + t * 16;

        // gather 16 feature rows (256B/row, coalesced float4)
        float4 g = make_float4(0.f, 0.f, 0.f, 0.f);
        const int pr = p0 + r;
        if (pr < P_PAIRS) {
            const int row = in_idx[k * P_PAIRS + pr];
            g = *(const float4*)(feat + (size_t)row * CH_IN + c4);
        }
        // out-row indices for this tile (threads 0..15), -1 marks tail pads
        int orow = -1;
        if (tid < 16 && (p0 + tid) < P_PAIRS)
            orow = out_idx[k * P_PAIRS + p0 + tid];

        __syncthreads();                       // prev tile's sA/sOut reads done
        *(float4*)&sA[r][c4] = g;
        if (tid < 16) sOut[tid] = orow;
        __syncthreads();

        // Hoist this half's 8 out-row indices into VGPRs now; the two
        // ds_load_b128 waits overlap with the WMMA stream below.
        const int4* op = (const int4*)&sOut[halfsel * 8];
        const int4 oa = op[0];
        const int4 ob = op[1];
        int orows[8] = { oa.x, oa.y, oa.z, oa.w, ob.x, ob.y, ob.z, ob.w };

        // 16 k-steps of V_WMMA_F32_16X16X4_F32 (uniform flow: EXEC all-1s)
        v8f acc = {};
#pragma unroll
        for (int s = 0; s < 16; ++s) {
            v2f a = *(const v2f*)&sA[m][4 * s + krow];
            acc = __builtin_amdgcn_wmma_f32_16x16x4_f32(
                false, a, false, bfrag[s], (short)0, acc, false, false);
        }

        // scatter-add: D VGPR r8 -> row M = r8 + halfsel*8, col n0 + m.
        // 32-bit element offsets (max 32M) -> saddr-form global atomics.
        if (full) {
            // guard-free: atomics issue back-to-back with full EXEC
#pragma unroll
            for (int r8 = 0; r8 < 8; ++r8) {
                const unsigned off = (unsigned)orows[r8] * CH_OUT + colofs;
                unsafeAtomicAdd(out + off, acc[r8]);
            }
        } else {
#pragma unroll
            for (int r8 = 0; r8 < 8; ++r8) {
                if (orows[r8] >= 0) {
                    const unsigned off = (unsigned)orows[r8] * CH_OUT + colofs;
                    unsafeAtomicAdd(out + off, acc[r8]);
                }
            }
        }
    }
}

// ---------------------------------------------------------------------------
// Kernel 2: per-channel sum / sumsq partial reduction (thread = channel)
// ---------------------------------------------------------------------------
__global__ __launch_bounds__(128) void bn_stats(
    const float* __restrict__ out, float* __restrict__ ws /* [256] */)
{
    const int c  = threadIdx.x;
    const int r0 = blockIdx.x * 512;
    float s = 0.f, s2 = 0.f;
#pragma unroll 4
    for (int i = 0; i < 512; ++i) {
        const int rr = r0 + i;
        if (rr < N_OUT_C) {
            const float v = out[(unsigned)rr * CH_OUT + c];
            s  += v;
            s2 += v * v;
        }
    }
    unsafeAtomicAdd(ws + c, s);
    unsafeAtomicAdd(ws + CH_OUT + c, s2);
}

// ---------------------------------------------------------------------------
// Kernel 3: BatchNorm (biased var) + ReLU, in place
// ---------------------------------------------------------------------------
__global__ __launch_bounds__(256) void bn_norm(
    float* __restrict__ out, const float* __restrict__ ws,
    const float* __restrict__ gamma, const float* __restrict__ beta)
{
    const unsigned i = blockIdx.x * 256u + threadIdx.x;
    const int   c     = (int)(i & (CH_OUT - 1));
    const float inv_n = 1.0f / (float)N_OUT_C;
    const float mean  = ws[c] * inv_n;
    const float var   = ws[CH_OUT + c] * inv_n - mean * mean;
    const float sc    = rsqrtf(var + BN_EPS) * gamma[c];
    const float v     = (out[i] - mean) * sc + beta[c];
    out[i] = v > 0.f ? v : 0.f;
}

// ---------------------------------------------------------------------------
extern "C" void kernel_launch(void* const* d_in, const int* in_sizes, int n_in,
                              void* d_out, int out_size, void* d_ws, size_t ws_size,
                              hipStream_t stream)
{
    const float* feat    = (const float*)d_in[0];   // [1e6, 64]
    const float* weight  = (const float*)d_in[1];   // [8, 64, 128]
    const float* gamma   = (const float*)d_in[2];   // [128]
    const float* beta    = (const float*)d_in[3];   // [128]
    const int*   in_idx  = (const int*)d_in[4];     // [8, 125000]
    const int*   out_idx = (const int*)d_in[5];     // [8, 125000]
    float*       out     = (float*)d_out;           // [250000, 128]
    float*       ws      = (float*)d_ws;            // 256 floats of stats

    // zero the accumulator and the stats scratch (capture-safe memset nodes)
    hipMemsetAsync(out, 0, (size_t)N_OUT_C * CH_OUT * sizeof(float), stream);
    hipMemsetAsync(ws, 0, 2 * CH_OUT * sizeof(float), stream);

    dim3 grid1((P_PAIRS + 127) / 128, K_OFF);
    spconv_gemm<<<grid1, 256, 0, stream>>>(feat, weight, in_idx, out_idx, out);

    bn_stats<<<(N_OUT_C + 511) / 512, 128, 0, stream>>>(out, ws);

    const unsigned total = N_OUT_C * CH_OUT;        // 32,000,000
    bn_norm<<<total / 256, 256, 0, stream>>>(out, ws, gamma, beta);
}